// GCODLoss_24223615550153
// MI455X (gfx1250) — compile-verified
//
#include <hip/hip_runtime.h>
#include <math.h>

#define LAMBDA_SMOOTHNESS 1.0f

typedef __attribute__((ext_vector_type(2))) float v2f;
typedef __attribute__((ext_vector_type(8))) float v8f;

// Wave32 all-reduce using V_WMMA_F32_16X16X4_F32.
// A = ones(16x4)  =>  D[i][j] = sum_k B[k][j]  (column sums, identical rows).
// Each lane contributes one value via B's first VGPR (second VGPR = 0).
// Across VGPR0's 32 lanes each column index N appears exactly twice, so
// sum_lanes(c[0]) == 2 * sum_lanes(v).  EXEC must be all ones at call site.
__device__ __forceinline__ float wave_allreduce_wmma(float v) {
  v2f a; a.x = 1.0f; a.y = 1.0f;   // all 16x4 slots = 1.0
  v2f b; b.x = v;    b.y = 0.0f;   // one partial per lane
  v8f c = {};
  c = __builtin_amdgcn_wmma_f32_16x16x4_f32(
      /*neg_a=*/false, a, /*neg_b=*/false, b,
      /*c_mod=*/(short)0, c, /*reuse_a=*/false, /*reuse_b=*/false);
  float s = c[0];
  s += __shfl_xor(s, 1, 32);
  s += __shfl_xor(s, 2, 32);
  s += __shfl_xor(s, 4, 32);
  s += __shfl_xor(s, 8, 32);
  s += __shfl_xor(s, 16, 32);
  return 0.5f * s;
}

__global__ void k_zero(float* __restrict__ p, int n) {
  int i = blockIdx.x * blockDim.x + threadIdx.x;
  if (i < n) p[i] = 0.0f;
}

// deg = segment_sum(within, row)   (row endpoint only, as in the reference)
__global__ void k_degree(const int* __restrict__ row, const int* __restrict__ col,
                         const int* __restrict__ batch, float* __restrict__ deg, int E) {
  int e = blockIdx.x * blockDim.x + threadIdx.x;
  if (e < E) {
    int r = row[e], c = col[e];
    if (batch[r] == batch[c]) atomicAdd(&deg[r], 1.0f);
  }
}

// deg -> deg^{-1/2} in place (0 where deg == 0)
__global__ void k_dinv(float* __restrict__ deg, int n) {
  int i = blockIdx.x * blockDim.x + threadIdx.x;
  if (i < n) {
    float d = deg[i];
    deg[i] = (d > 0.0f) ? (1.0f / sqrtf(d)) : 0.0f;
  }
}

// One wave32 per edge (grid-stride). Wave loads two contiguous 1KB rows
// (float4, fully coalesced), accumulates norm * ||x_r - x_c||^2 into a
// per-lane partial, then WMMA wave-reduce + one atomic per wave.
__global__ __launch_bounds__(256)
void k_energy(const float* __restrict__ x,
              const int* __restrict__ row, const int* __restrict__ col,
              const int* __restrict__ batch, const float* __restrict__ dinv,
              float* __restrict__ energy, int E, int D) {
  const int lane  = threadIdx.x & 31;
  const int wave  = threadIdx.x >> 5;
  const int wpb   = blockDim.x >> 5;
  const long long gwave  = (long long)blockIdx.x * wpb + wave;
  const long long nwaves = (long long)gridDim.x * wpb;
  const int nv4 = D >> 2;

  float wsum = 0.0f;
  for (long long e = gwave; e < (long long)E; e += nwaves) {
    int r = row[e];          // wave-uniform
    int c = col[e];
    if (batch[r] != batch[c]) continue;          // uniform branch: EXEC stays full
    float nrm = dinv[r] * dinv[c];
    if (nrm == 0.0f) continue;

    const float4* __restrict__ xr = (const float4*)(x + (size_t)r * D);
    const float4* __restrict__ xc = (const float4*)(x + (size_t)c * D);
    float acc = 0.0f;
    for (int i = lane; i < nv4; i += 32) {       // D=256 -> 2 iterations/lane
      float4 a = xr[i];
      float4 b = xc[i];
      float d0 = a.x - b.x, d1 = a.y - b.y, d2 = a.z - b.z, d3 = a.w - b.w;
      acc = fmaf(d0, d0, acc);
      acc = fmaf(d1, d1, acc);
      acc = fmaf(d2, d2, acc);
      acc = fmaf(d3, d3, acc);
    }
    wsum = fmaf(nrm, acc, wsum);
  }

  float tot = wave_allreduce_wmma(wsum);         // EXEC all ones here
  if (lane == 0) atomicAdd(energy, tot);
}

// CE over G graphs (one thread per graph, G == blockDim.x, multiple of 32),
// WMMA wave-reduce, then combine with smoothness term.
__global__ void k_final(const float* __restrict__ logits, const int* __restrict__ labels,
                        const int* __restrict__ batch, int n_nodes, int n_classes,
                        const float* __restrict__ energy, float* __restrict__ out) {
  __shared__ float warp_part[32];
  const int t    = threadIdx.x;
  const int G    = blockDim.x;
  const int lane = t & 31;
  const int wave = t >> 5;

  const float* lg = logits + (size_t)t * n_classes;
  float m = lg[0];
  for (int j = 1; j < n_classes; ++j) m = fmaxf(m, lg[j]);
  float se = 0.0f;
  for (int j = 0; j < n_classes; ++j) se += expf(lg[j] - m);
  float lse  = m + logf(se);
  float loss = lse - lg[labels[t]];
  float part = loss / (float)G;

  float wtot = wave_allreduce_wmma(part);        // all G threads active
  if (lane == 0) warp_part[wave] = wtot;
  __syncthreads();

  if (t == 0) {
    float ce = 0.0f;
    int nw = G >> 5;
    for (int w = 0; w < nw; ++w) ce += warp_part[w];
    float num_graphs = (float)(batch[n_nodes - 1] + 1);   // batch is sorted
    out[0] = ce + LAMBDA_SMOOTHNESS * (energy[0] / num_graphs);
  }
}

extern "C" void kernel_launch(void* const* d_in, const int* in_sizes, int n_in,
                              void* d_out, int out_size, void* d_ws, size_t ws_size,
                              hipStream_t stream) {
  const float* logits = (const float*)d_in[0];   // [G, C]
  const int*   labels = (const int*)d_in[1];     // [G]
  const float* x      = (const float*)d_in[2];   // [N, D]
  const int*   edge   = (const int*)d_in[3];     // [2, E]
  const int*   batch  = (const int*)d_in[4];     // [N], sorted

  const int G = in_sizes[1];
  const int C = in_sizes[0] / G;
  const int N = in_sizes[4];
  const int E = in_sizes[3] / 2;
  const int D = in_sizes[2] / N;

  float* ws_f   = (float*)d_ws;
  float* energy = ws_f;        // 1 float
  float* deg    = ws_f + 64;   // N floats (256B-aligned offset)

  const int* row = edge;
  const int* col = edge + E;

  const int nz = N + 64;
  k_zero<<<(nz + 255) / 256, 256, 0, stream>>>(ws_f, nz);
  k_degree<<<(E + 255) / 256, 256, 0, stream>>>(row, col, batch, deg, E);
  k_dinv<<<(N + 255) / 256, 256, 0, stream>>>(deg, N);
  k_energy<<<4096, 256, 0, stream>>>(x, row, col, batch, deg, energy, E, D);
  k_final<<<1, G, 0, stream>>>(logits, labels, batch, N, C, energy, (float*)d_out);
}